// GAT_22058952032367
// MI455X (gfx1250) — compile-verified
//
#include <hip/hip_runtime.h>
#include <hip/hip_bf16.h>

typedef __attribute__((ext_vector_type(2))) float v2f;
typedef __attribute__((ext_vector_type(8))) float v8f;

#define NEG_SLOPE 0.2f

// ---------------- order-preserving float<->uint encoding for atomic max ----
__device__ __forceinline__ unsigned f2ord(float f) {
    unsigned b = __float_as_uint(f);
    return (b & 0x80000000u) ? ~b : (b | 0x80000000u);
}
__device__ __forceinline__ float ord2f(unsigned u) {
    unsigned b = (u & 0x80000000u) ? (u & 0x7FFFFFFFu) : ~u;
    return __uint_as_float(b);
}
#define ORD_NEG_INF 0x007FFFFFu  // f2ord(-inf)

// ---------------- edge (src,dst) with appended self-loops ------------------
__device__ __forceinline__ void edge_sd(const int* ei, int E0, int e, int& s, int& d) {
    if (e < E0) { s = ei[e]; d = ei[E0 + e]; }
    else        { s = e - E0; d = s; }
}

// ===========================================================================
// WMMA fp32 GEMM: C[M,Ntot] = A[M,K] @ B[K,Ntot].  K,Ntot multiples of 64.
// 256 threads (8 wave32), 64x64 tile, LDS-staged, V_WMMA_F32_16X16X4_F32.
// B is stored in LDS pre-paired (Bp[k/2][2n + (k&1)]) so each lane's
// B-fragment is one aligned ds_load_b64 (no v_mov shuffling before WMMA).
// ===========================================================================
__global__ void gat_wmma_gemm(const float* __restrict__ A,
                              const float* __restrict__ B,
                              float* __restrict__ C,
                              int M, int Ntot, int K) {
    __shared__ float As[64][68];    // [m][k]  (68-stride: conflict-free b64 reads)
    __shared__ float Bp[32][132];   // [k/2][2n + (k&1)], n in 0..63

    const int tid   = threadIdx.x;
    const int lane  = tid & 31;
    const int wave  = tid >> 5;
    const int mBase = blockIdx.y * 64;
    const int nBase = blockIdx.x * 64;

    const int ms   = (wave & 3) * 16;   // wave's M sub-tile origin in tile
    const int ns   = (wave >> 2) * 32;  // wave's N half (two 16-wide subtiles)
    const int mrow = ms + (lane & 15);  // A-fragment row for this lane
    const int khi  = (lane >> 4) * 2;   // fragment K offset (ISA 16x4 layout)
    const int ncol = lane & 15;

    v8f acc0 = {}; v8f acc1 = {};

    for (int kBase = 0; kBase < K; kBase += 64) {
        // cooperative tile load: 1024 float4 slots, 4 per thread
#pragma unroll
        for (int i = 0; i < 4; ++i) {
            int s  = tid + i * 256;
            int r  = s >> 4;             // 0..63
            int c4 = (s & 15) << 2;      // 0,4,...,60
            float4 va = make_float4(0.f, 0.f, 0.f, 0.f);
            int g = mBase + r;
            if (g < M) va = *(const float4*)(A + (size_t)g * K + kBase + c4);
            *(float4*)&As[r][c4] = va;
            float4 vb = *(const float4*)(B + (size_t)(kBase + r) * Ntot + nBase + c4);
            float* bp = &Bp[r >> 1][r & 1];   // element (k=r, n=c) -> [r/2][2c+(r&1)]
            bp[2 * c4 + 0] = vb.x;
            bp[2 * c4 + 2] = vb.y;
            bp[2 * c4 + 4] = vb.z;
            bp[2 * c4 + 6] = vb.w;
        }
        __syncthreads();

#pragma unroll
        for (int kk = 0; kk < 64; kk += 4) {
            const int p2 = (kk + khi) >> 1;   // paired-row index (kk+khi even)
            v2f a, b0, b1v;
            a[0]   = As[mrow][kk + khi];
            a[1]   = As[mrow][kk + khi + 1];
            b0     = *(const v2f*)&Bp[p2][2 * (ns + ncol)];        // ds_load_b64
            b1v    = *(const v2f*)&Bp[p2][2 * (ns + 16 + ncol)];   // ds_load_b64
            acc0 = __builtin_amdgcn_wmma_f32_16x16x4_f32(
                false, a, false, b0, (short)0, acc0, false, false);
            acc1 = __builtin_amdgcn_wmma_f32_16x16x4_f32(
                false, a, false, b1v, (short)0, acc1, false, false);
        }
        __syncthreads();
    }

    // store: VGPR r -> row (ms + r) lanes 0-15, (ms + r + 8) lanes 16-31
    const int orow = mBase + ms + ((lane >> 4) << 3);
    const int ocol = nBase + ns + ncol;
#pragma unroll
    for (int r = 0; r < 8; ++r) {
        int g = orow + r;
        if (g < M) {
            C[(size_t)g * Ntot + ocol]      = acc0[r];
            C[(size_t)g * Ntot + ocol + 16] = acc1[r];
        }
    }
}

// ===========================================================================
// Layer-1 helpers
// ===========================================================================
__global__ void scores1_kernel(const float* __restrict__ h1,
                               const float* __restrict__ att_s,
                               const float* __restrict__ att_d,
                               float* __restrict__ as1, float* __restrict__ ad1,
                               int Nn) {
    int t = blockIdx.x * blockDim.x + threadIdx.x;
    if (t >= Nn * 8) return;
    int n = t >> 3, h = t & 7;
    const float* row = h1 + (size_t)n * 256 + h * 32;
    const float* ws = att_s + h * 32;
    const float* wd = att_d + h * 32;
    float ss = 0.f, sd = 0.f;
#pragma unroll
    for (int c = 0; c < 32; ++c) { ss += row[c] * ws[c]; sd += row[c] * wd[c]; }
    as1[t] = ss; ad1[t] = sd;
}

__global__ void init1_kernel(float* __restrict__ num1, float* __restrict__ den1,
                             unsigned* __restrict__ amax1, int Nn) {
    int t = blockIdx.x * blockDim.x + threadIdx.x;
    if (t < Nn * 256) num1[t] = 0.f;
    if (t < Nn * 8) { den1[t] = 0.f; amax1[t] = ORD_NEG_INF; }
}

__global__ void edge_max1_kernel(const int* __restrict__ ei, int E0, int Nn,
                                 const float* __restrict__ as1,
                                 const float* __restrict__ ad1,
                                 unsigned* __restrict__ amax1) {
    int t = blockIdx.x * blockDim.x + threadIdx.x;
    int ET = E0 + Nn;
    if (t >= ET * 8) return;
    int e = t >> 3, h = t & 7;
    int s, d; edge_sd(ei, E0, e, s, d);
    float lg = as1[s * 8 + h] + ad1[d * 8 + h];
    lg = lg > 0.f ? lg : NEG_SLOPE * lg;
    atomicMax(&amax1[d * 8 + h], f2ord(lg));
}

// one wave32 per edge; lanes = channels -> coalesced 128B gathers & atomics
__global__ void edge_acc1_kernel(const int* __restrict__ ei, int E0, int Nn,
                                 const float* __restrict__ as1,
                                 const float* __restrict__ ad1,
                                 const unsigned* __restrict__ amax1,
                                 const float* __restrict__ h1,
                                 float* __restrict__ den1,
                                 float* __restrict__ num1) {
    int gtid = blockIdx.x * blockDim.x + threadIdx.x;
    int e    = gtid >> 5;              // one wave per edge
    int lane = threadIdx.x & 31;
    if (e >= E0 + Nn) return;          // wave-uniform exit
    int s, d; edge_sd(ei, E0, e, s, d);
    const float* hrow = h1 + (size_t)s * 256;
    float*       orow = num1 + (size_t)d * 256;
    __builtin_prefetch(hrow, 0, 1);    // global_prefetch_b8
#pragma unroll
    for (int h = 0; h < 8; ++h) {
        float lg = as1[s * 8 + h] + ad1[d * 8 + h];   // wave-uniform
        lg = lg > 0.f ? lg : NEG_SLOPE * lg;
        float p = expf(lg - ord2f(amax1[d * 8 + h]));
        if (lane == 0) atomicAdd(&den1[d * 8 + h], p);
        atomicAdd(&orow[h * 32 + lane], hrow[h * 32 + lane] * p);
    }
}

__global__ void norm_elu1_kernel(const float* __restrict__ num1,
                                 const float* __restrict__ den1,
                                 const float* __restrict__ b1,
                                 float* __restrict__ hout, int Nn) {
    int t = blockIdx.x * blockDim.x + threadIdx.x;
    if (t >= Nn * 256) return;
    int n = t >> 8, idx = t & 255, h = idx >> 5;
    float v = num1[t] / (den1[n * 8 + h] + 1e-16f) + b1[idx];
    hout[t] = v > 0.f ? v : (expf(v) - 1.f);
}

// ===========================================================================
// Layer-2 helpers (H = 1, C = 64)
// ===========================================================================
__global__ void scores2_kernel(const float* __restrict__ h2,
                               const float* __restrict__ att_s,
                               const float* __restrict__ att_d,
                               float* __restrict__ as2, float* __restrict__ ad2,
                               int Nn) {
    int n = blockIdx.x * blockDim.x + threadIdx.x;
    if (n >= Nn) return;
    const float* row = h2 + (size_t)n * 64;
    float ss = 0.f, sd = 0.f;
#pragma unroll
    for (int c = 0; c < 64; ++c) { ss += row[c] * att_s[c]; sd += row[c] * att_d[c]; }
    as2[n] = ss; ad2[n] = sd;
}

__global__ void init2_kernel(float* __restrict__ out, float* __restrict__ den2,
                             unsigned* __restrict__ amax2, int Nn) {
    int t = blockIdx.x * blockDim.x + threadIdx.x;
    if (t < Nn * 64) out[t] = 0.f;
    if (t < Nn) { den2[t] = 0.f; amax2[t] = ORD_NEG_INF; }
}

__global__ void edge_max2_kernel(const int* __restrict__ ei, int E0, int Nn,
                                 const float* __restrict__ as2,
                                 const float* __restrict__ ad2,
                                 unsigned* __restrict__ amax2) {
    int e = blockIdx.x * blockDim.x + threadIdx.x;
    if (e >= E0 + Nn) return;
    int s, d; edge_sd(ei, E0, e, s, d);
    float lg = as2[s] + ad2[d];
    lg = lg > 0.f ? lg : NEG_SLOPE * lg;
    atomicMax(&amax2[d], f2ord(lg));
}

// one wave32 per edge; 64 channels = 2 coalesced strips
__global__ void edge_acc2_kernel(const int* __restrict__ ei, int E0, int Nn,
                                 const float* __restrict__ as2,
                                 const float* __restrict__ ad2,
                                 const unsigned* __restrict__ amax2,
                                 const float* __restrict__ h2,
                                 float* __restrict__ den2,
                                 float* __restrict__ out) {
    int gtid = blockIdx.x * blockDim.x + threadIdx.x;
    int e    = gtid >> 5;
    int lane = threadIdx.x & 31;
    if (e >= E0 + Nn) return;          // wave-uniform exit
    int s, d; edge_sd(ei, E0, e, s, d);
    float lg = as2[s] + ad2[d];
    lg = lg > 0.f ? lg : NEG_SLOPE * lg;
    float p = expf(lg - ord2f(amax2[d]));
    if (lane == 0) atomicAdd(&den2[d], p);
    const float* hrow = h2 + (size_t)s * 64;
    float*       orow = out + (size_t)d * 64;
    __builtin_prefetch(hrow, 0, 1);
    atomicAdd(&orow[lane],      hrow[lane]      * p);
    atomicAdd(&orow[32 + lane], hrow[32 + lane] * p);
}

__global__ void final_kernel(float* __restrict__ out,
                             const float* __restrict__ den2,
                             const float* __restrict__ b2, int Nn) {
    int t = blockIdx.x * blockDim.x + threadIdx.x;
    if (t >= Nn * 64) return;
    int n = t >> 6, c = t & 63;
    out[t] = out[t] / (den2[n] + 1e-16f) + b2[c];   // mean over 1 head == identity
}

// ===========================================================================
extern "C" void kernel_launch(void* const* d_in, const int* in_sizes, int n_in,
                              void* d_out, int out_size, void* d_ws, size_t ws_size,
                              hipStream_t stream) {
    const float* x        = (const float*)d_in[0];
    const int*   ei       = (const int*)d_in[1];
    const float* W1       = (const float*)d_in[2];
    const float* att_src1 = (const float*)d_in[3];
    const float* att_dst1 = (const float*)d_in[4];
    const float* b1       = (const float*)d_in[5];
    const float* W2       = (const float*)d_in[6];
    const float* att_src2 = (const float*)d_in[7];
    const float* att_dst2 = (const float*)d_in[8];
    const float* b2       = (const float*)d_in[9];
    float* out = (float*)d_out;

    const int Nn = in_sizes[0] / 256;   // IN_C = 256
    const int E0 = in_sizes[1] / 2;
    const int ET = E0 + Nn;

    // workspace layout (floats)
    float*    h1    = (float*)d_ws;                 // N*256 (reused in-place for ELU output)
    float*    num1  = h1 + (size_t)Nn * 256;        // N*256 ; aliased by h2 after layer 1
    float*    as1   = num1 + (size_t)Nn * 256;      // N*8
    float*    ad1   = as1 + (size_t)Nn * 8;         // N*8
    unsigned* amax1 = (unsigned*)(ad1 + (size_t)Nn * 8); // N*8
    float*    den1  = (float*)(amax1 + (size_t)Nn * 8);  // N*8
    // layer-2 aliases (regions dead after layer-1 edge passes)
    float*    h2    = num1;                         // N*64
    float*    as2   = as1;                          // N
    float*    ad2   = as1 + Nn;                     // N
    unsigned* amax2 = (unsigned*)(as1 + 2 * (size_t)Nn); // N
    float*    den2  = as1 + 3 * (size_t)Nn;         // N

    const int TB = 256;
    dim3 g1((256 + 63) / 64, (Nn + 63) / 64);   // GEMM1: Ntot=256
    dim3 g2((64 + 63) / 64, (Nn + 63) / 64);    // GEMM2: Ntot=64

    // ---- layer 1 ----
    gat_wmma_gemm<<<g1, TB, 0, stream>>>(x, W1, h1, Nn, 256, 256);
    scores1_kernel<<<(Nn * 8 + TB - 1) / TB, TB, 0, stream>>>(h1, att_src1, att_dst1, as1, ad1, Nn);
    init1_kernel<<<(Nn * 256 + TB - 1) / TB, TB, 0, stream>>>(num1, den1, amax1, Nn);
    edge_max1_kernel<<<(ET * 8 + TB - 1) / TB, TB, 0, stream>>>(ei, E0, Nn, as1, ad1, amax1);
    edge_acc1_kernel<<<(ET + 7) / 8, TB, 0, stream>>>(ei, E0, Nn, as1, ad1, amax1,
                                                      h1, den1, num1);   // 1 wave / edge
    norm_elu1_kernel<<<(Nn * 256 + TB - 1) / TB, TB, 0, stream>>>(num1, den1, b1, h1, Nn);

    // ---- layer 2 ----
    gat_wmma_gemm<<<g2, TB, 0, stream>>>(h1, W2, h2, Nn, 64, 256);
    scores2_kernel<<<(Nn + TB - 1) / TB, TB, 0, stream>>>(h2, att_src2, att_dst2, as2, ad2, Nn);
    init2_kernel<<<(Nn * 64 + TB - 1) / TB, TB, 0, stream>>>(out, den2, amax2, Nn);
    edge_max2_kernel<<<(ET + TB - 1) / TB, TB, 0, stream>>>(ei, E0, Nn, as2, ad2, amax2);
    edge_acc2_kernel<<<(ET + 7) / 8, TB, 0, stream>>>(ei, E0, Nn, as2, ad2, amax2,
                                                      h2, den2, out);    // 1 wave / edge
    final_kernel<<<(Nn * 64 + TB - 1) / TB, TB, 0, stream>>>(out, den2, b2, Nn);
}